// ProjectionLoss_6262062318053
// MI455X (gfx1250) — compile-verified
//
#include <hip/hip_runtime.h>
#include <math.h>

// Problem constants from the reference
#define KNN   8
#define BATCH 4
#define NPTS  8192   // gts per batch
#define MPTS  8192   // preds per batch

typedef float v2f __attribute__((ext_vector_type(2)));
typedef float v8f __attribute__((ext_vector_type(8)));

static constexpr float INV_SIGMA_P2    = 1.0f / (0.03f * 0.03f);  // 1/sigma_p^2
static constexpr float INV_ANGLE_DENOM = 29.34742503084443f;      // 1/(1-cos(pi/12))

// One wave owns 32 query rows. Per 16-column chunk of gt points it runs two
// V_WMMA_F32_16X16X4_F32 ops computing (-2 q.p + |p|^2) for a 32x16 tile,
// transposes the tile through LDS, and each lane maintains a register-resident
// sorted top-8 (value+index) for its row.
__global__ __launch_bounds__(256)
void knn_projection_kernel(const float* __restrict__ preds,
                           const float* __restrict__ gts,
                           const float* __restrict__ normals,
                           float* __restrict__ partials)
{
    __shared__ float tile[8][32][16];   // per-wave 32x16 transpose staging (16 KB)
    __shared__ float wavesum[8];

    const int lane = threadIdx.x & 31;
    const int wave = threadIdx.x >> 5;
    const int b    = blockIdx.y;
    const int m0   = blockIdx.x * 256 + wave * 32;   // first query row of this wave

    const float* predsB = preds   + (size_t)b * MPTS * 3;
    const float* gtsB   = gts     + (size_t)b * NPTS * 3;
    const float* nrmB   = normals + (size_t)b * NPTS * 3;

    const int  r  = lane & 15;
    const bool lo = lane < 16;

    // ---- A fragments: two 16x4 query tiles, K-dim = (qx, qy, qz, 1)
    // ISA layout: lanes 0-15 hold K=0 (v0) / K=1 (v1); lanes 16-31 hold K=2 / K=3.
    v2f a0, a1;
    {
        const float* q0 = predsB + (size_t)(m0 + r) * 3;
        float x0 = q0[0], y0 = q0[1], z0 = q0[2];
        a0.x = lo ? x0 : z0;
        a0.y = lo ? y0 : 1.0f;
        const float* q1 = predsB + (size_t)(m0 + 16 + r) * 3;
        float x1 = q1[0], y1 = q1[1], z1 = q1[2];
        a1.x = lo ? x1 : z1;
        a1.y = lo ? y1 : 1.0f;
    }

    // Per-lane sorted (ascending) top-8 of (-2 q.p + |p|^2) for row m0+lane
    float bestD[KNN];
    int   bestI[KNN];
#pragma unroll
    for (int k = 0; k < KNN; ++k) { bestD[k] = 3.4e38f; bestI[k] = 0; }

    for (int n0 = 0; n0 < NPTS; n0 += 16) {
        // ---- B fragment: 4x16 gt tile, K-dim = (-2gx, -2gy, -2gz, |g|^2)
        const float* g = gtsB + (size_t)(n0 + r) * 3;
        float gx = g[0], gy = g[1], gz = g[2];
        float g2 = gx * gx + gy * gy + gz * gz;
        v2f bb;
        bb.x = lo ? (-2.0f * gx) : (-2.0f * gz);
        bb.y = lo ? (-2.0f * gy) : g2;

        if (n0 + 16 < NPTS) {
            __builtin_prefetch(gtsB + (size_t)(n0 + 16 + r) * 3, 0, 3);
        }

        v8f cz = {};
        v8f c0 = __builtin_amdgcn_wmma_f32_16x16x4_f32(
            false, a0, false, bb, (short)0, cz, false, false);
        v8f c1 = __builtin_amdgcn_wmma_f32_16x16x4_f32(
            false, a1, false, bb, (short)0, cz, false, false);

        // ---- scatter C tiles to LDS (row-major [32 rows][16 cols] per wave)
        // C layout: VGPR v -> (row v, col lane) lanes 0-15; (row v+8, col lane-16) lanes 16-31.
#pragma unroll
        for (int v = 0; v < 8; ++v) {
            int row = lo ? v : (v + 8);
            tile[wave][row][r]      = c0[v];
            tile[wave][row + 16][r] = c1[v];
        }
        // wave-synchronous: same-wave LDS ops are in-order; wait before cross-lane read
        asm volatile("s_wait_dscnt 0" ::: "memory");

        // ---- each lane scans its own row's 16 candidates
        const float4* rowp = (const float4*)&tile[wave][lane][0];
#pragma unroll
        for (int jj = 0; jj < 4; ++jj) {
            float4 q4 = rowp[jj];
            float cand[4] = { q4.x, q4.y, q4.z, q4.w };
#pragma unroll
            for (int e = 0; e < 4; ++e) {
                float d   = cand[e];
                int   idx = n0 + jj * 4 + e;
                if (d < bestD[KNN - 1]) {        // usually fails -> cheap
                    bestD[KNN - 1] = d;
                    bestI[KNN - 1] = idx;
#pragma unroll
                    for (int k = KNN - 1; k >= 1; --k) {
                        if (bestD[k] < bestD[k - 1]) {
                            float td = bestD[k]; bestD[k] = bestD[k - 1]; bestD[k - 1] = td;
                            int   ti = bestI[k]; bestI[k] = bestI[k - 1]; bestI[k - 1] = ti;
                        }
                    }
                }
            }
        }
        // protect LDS WAR vs. next iteration's stores
        asm volatile("s_wait_dscnt 0" ::: "memory");
    }

    // ---- phase 2: weighted projection for row q = m0 + lane
    const float* q = predsB + (size_t)(m0 + lane) * 3;
    float qx = q[0], qy = q[1], qz = q[2];
    float q2 = qx * qx + qy * qy + qz * qz;

    const float* en = nrmB + (size_t)bestI[0] * 3;   // normal of nearest neighbor
    float ex = en[0], ey = en[1], ez = en[2];

    float wsum = 0.0f, psum = 0.0f;
#pragma unroll
    for (int k = 0; k < KNN; ++k) {
        int ik = bestI[k];
        const float* nk = nrmB + (size_t)ik * 3;
        const float* pk = gtsB + (size_t)ik * 3;
        float nx = nk[0], ny = nk[1], nz = nk[2];
        float px = pk[0], py = pk[1], pz = pk[2];

        float dist  = bestD[k] + q2;                       // true squared distance
        float dw    = __expf(-dist * INV_SIGMA_P2);
        float inner = nx * ex + ny * ey + nz * ez;
        float aw    = __expf(-(1.0f - inner) * INV_ANGLE_DENOM);
        float w     = dw * aw;

        float dx = qx - px, dy = qy - py, dz = qz - pz;
        float proj = fabsf(dx * nx + dy * ny + dz * nz);
        wsum += w;
        psum += proj * w;
    }
    float disp = psum / wsum;

    // ---- deterministic reduction: wave -> block -> per-block partial
#pragma unroll
    for (int off = 16; off >= 1; off >>= 1)
        disp += __shfl_xor(disp, off, 32);
    if (lane == 0) wavesum[wave] = disp;
    __syncthreads();
    if (threadIdx.x == 0) {
        float s = 0.0f;
#pragma unroll
        for (int wv = 0; wv < 8; ++wv) s += wavesum[wv];
        partials[blockIdx.y * gridDim.x + blockIdx.x] = s;
    }
}

// Fixed-order tree reduction of the 128 per-block partials -> scalar output.
__global__ void final_reduce_kernel(const float* __restrict__ partials,
                                    float* __restrict__ out, int n)
{
    __shared__ float buf[128];
    int t = threadIdx.x;
    buf[t] = (t < n) ? partials[t] : 0.0f;
    __syncthreads();
    for (int s = 64; s >= 1; s >>= 1) {
        if (t < s) buf[t] += buf[t + s];
        __syncthreads();
    }
    if (t == 0) out[0] = buf[0];
}

extern "C" void kernel_launch(void* const* d_in, const int* in_sizes, int n_in,
                              void* d_out, int out_size, void* d_ws, size_t ws_size,
                              hipStream_t stream)
{
    const float* preds   = (const float*)d_in[0];   // (B, M, 3) f32
    const float* gts     = (const float*)d_in[1];   // (B, N, 3) f32
    const float* normals = (const float*)d_in[2];   // (B, N, 3) f32
    float* out      = (float*)d_out;                // scalar
    float* partials = (float*)d_ws;                 // 128 floats of scratch

    dim3 grid(MPTS / 256, BATCH, 1);                // 32 x 4 = 128 blocks
    dim3 block(256, 1, 1);                          // 8 waves/block (wave32)
    knn_projection_kernel<<<grid, block, 0, stream>>>(preds, gts, normals, partials);
    final_reduce_kernel<<<1, 128, 0, stream>>>(partials, out, grid.x * grid.y);
}